// Memory_76957224010242
// MI455X (gfx1250) — compile-verified
//
#include <hip/hip_runtime.h>
#include <math.h>

// MI455X / gfx1250, wave32. fp32 GEMMs via V_WMMA_F32_16X16X4_F32.
typedef float v2f __attribute__((ext_vector_type(2)));
typedef float v8f __attribute__((ext_vector_type(8)));

#define BSZ  512    // batch
#define HDIM 2048   // latent dim (H)
#define DDIM 4096   // data dim (D)

// ---------------------------------------------------------------------------
// Kernel 1: diff = x - r @ W^T - c      (M=512, N=4096, K=2048)
//   A = r   (M x K, row-major, K contiguous)
//   B[k][n] = W[n*K + k]  (W is D x H row-major == N x K row-major)
// Also emits per-block partial sums of diff^2.
// Block tile 64(M) x 128(N), 256 threads = 8 waves, wave tile 32x32 (2x2 WMMA).
// ---------------------------------------------------------------------------
__global__ __launch_bounds__(256) void fwd_kernel(
    const float* __restrict__ x, const float* __restrict__ W,
    const float* __restrict__ cvec, const float* __restrict__ r,
    float* __restrict__ diff, float* __restrict__ P1)
{
  __shared__ float As[64 * 20];    // pad 20: conflict-free, 16B-aligned rows
  __shared__ float Bs[128 * 20];
  __shared__ float red[256];

  const int tid  = threadIdx.x;
  const int lane = tid & 31;
  const int wave = tid >> 5;
  const int hl   = lane >> 4;      // half-wave
  const int lr   = lane & 15;
  const int wm   = wave >> 2;      // 0..1
  const int wn   = wave & 3;       // 0..3
  const int m0   = blockIdx.y * 64;
  const int n0   = blockIdx.x * 128;

  v8f acc[2][2] = {};

  const int arow = tid >> 2, aseg = (tid & 3) * 4;   // A: 64 rows x 16
  const int brow = tid >> 1, bseg = (tid & 1) * 8;   // B: 128 rows x 16

  for (int k0 = 0; k0 < HDIM; k0 += 16) {
    float4 av  = *(const float4*)(r + (size_t)(m0 + arow) * HDIM + k0 + aseg);
    float4 bv0 = *(const float4*)(W + (size_t)(n0 + brow) * HDIM + k0 + bseg);
    float4 bv1 = *(const float4*)(W + (size_t)(n0 + brow) * HDIM + k0 + bseg + 4);
    __syncthreads();
    *(float4*)&As[arow * 20 + aseg]     = av;
    *(float4*)&Bs[brow * 20 + bseg]     = bv0;
    *(float4*)&Bs[brow * 20 + bseg + 4] = bv1;
    __syncthreads();
#pragma unroll
    for (int kk = 0; kk < 4; ++kk) {
      const int kc = kk * 4 + hl * 2;
      v2f a0 = *(const v2f*)&As[(wm * 32 +      lr) * 20 + kc];
      v2f a1 = *(const v2f*)&As[(wm * 32 + 16 + lr) * 20 + kc];
      v2f b0 = *(const v2f*)&Bs[(wn * 32 +      lr) * 20 + kc];
      v2f b1 = *(const v2f*)&Bs[(wn * 32 + 16 + lr) * 20 + kc];
      acc[0][0] = __builtin_amdgcn_wmma_f32_16x16x4_f32(false, a0, false, b0, (short)0, acc[0][0], false, false);
      acc[0][1] = __builtin_amdgcn_wmma_f32_16x16x4_f32(false, a0, false, b1, (short)0, acc[0][1], false, false);
      acc[1][0] = __builtin_amdgcn_wmma_f32_16x16x4_f32(false, a1, false, b0, (short)0, acc[1][0], false, false);
      acc[1][1] = __builtin_amdgcn_wmma_f32_16x16x4_f32(false, a1, false, b1, (short)0, acc[1][1], false, false);
    }
  }

  // Epilogue: diff = x - acc - c ; accumulate diff^2
  float lsum = 0.f;
#pragma unroll
  for (int i = 0; i < 2; ++i)
#pragma unroll
    for (int j = 0; j < 2; ++j)
#pragma unroll
      for (int v = 0; v < 8; ++v) {
        const int gm = m0 + wm * 32 + i * 16 + v + hl * 8;
        const int gn = n0 + wn * 32 + j * 16 + lr;
        const size_t idx = (size_t)gm * DDIM + gn;
        const float d = x[idx] - acc[i][j][v] - cvec[gn];
        diff[idx] = d;
        lsum += d * d;
      }

  red[tid] = lsum;
  __syncthreads();
  for (int s = 128; s > 0; s >>= 1) {
    if (tid < s) red[tid] += red[tid + s];
    __syncthreads();
  }
  if (tid == 0) P1[blockIdx.y * gridDim.x + blockIdx.x] = red[0];
}

// ---------------------------------------------------------------------------
// Kernel 2: G = diff @ W  (M=512, N=2048, K=4096), fused Adam update of r.
//   A = diff (M x K row-major);  B[k][n] = W[k*H + n] (row-major K x N)
// grad = (2/B)*(pr - G); Adam step; emits per-block Σpr^2 and Σ‖Δr‖^2.
// ---------------------------------------------------------------------------
__global__ __launch_bounds__(256) void grad_adam_kernel(
    const float* __restrict__ diff, const float* __restrict__ W,
    const float* __restrict__ bvec, float* __restrict__ r,
    float* __restrict__ mst, float* __restrict__ vst,
    float* __restrict__ P2, float* __restrict__ P3,
    const int* __restrict__ done, float inv1, float inv2)
{
  __shared__ float As[64 * 20];
  __shared__ float Bs[16 * 136];   // 136: half-wave fragment reads hit disjoint banks
  __shared__ float red[256];

  const int tid  = threadIdx.x;
  const int lane = tid & 31;
  const int wave = tid >> 5;
  const int hl   = lane >> 4;
  const int lr   = lane & 15;
  const int wm   = wave >> 2;
  const int wn   = wave & 3;
  const int m0   = blockIdx.y * 64;
  const int n0   = blockIdx.x * 128;

  v8f acc[2][2] = {};

  const int arow = tid >> 2, aseg = (tid & 3) * 4;    // A: 64 x 16
  const int brow = tid >> 4, bseg = (tid & 15) * 8;   // B: 16 x 128

  for (int k0 = 0; k0 < DDIM; k0 += 16) {
    float4 av  = *(const float4*)(diff + (size_t)(m0 + arow) * DDIM + k0 + aseg);
    float4 bv0 = *(const float4*)(W + (size_t)(k0 + brow) * HDIM + n0 + bseg);
    float4 bv1 = *(const float4*)(W + (size_t)(k0 + brow) * HDIM + n0 + bseg + 4);
    __syncthreads();
    *(float4*)&As[arow * 20 + aseg]       = av;
    *(float4*)&Bs[brow * 136 + bseg]      = bv0;
    *(float4*)&Bs[brow * 136 + bseg + 4]  = bv1;
    __syncthreads();
#pragma unroll
    for (int kk = 0; kk < 4; ++kk) {
      const int kc = kk * 4 + hl * 2;
      v2f a0 = *(const v2f*)&As[(wm * 32 +      lr) * 20 + kc];
      v2f a1 = *(const v2f*)&As[(wm * 32 + 16 + lr) * 20 + kc];
      v2f b0, b1;
      b0[0] = Bs[(kc    ) * 136 + wn * 32 +      lr];
      b0[1] = Bs[(kc + 1) * 136 + wn * 32 +      lr];
      b1[0] = Bs[(kc    ) * 136 + wn * 32 + 16 + lr];
      b1[1] = Bs[(kc + 1) * 136 + wn * 32 + 16 + lr];
      acc[0][0] = __builtin_amdgcn_wmma_f32_16x16x4_f32(false, a0, false, b0, (short)0, acc[0][0], false, false);
      acc[0][1] = __builtin_amdgcn_wmma_f32_16x16x4_f32(false, a0, false, b1, (short)0, acc[0][1], false, false);
      acc[1][0] = __builtin_amdgcn_wmma_f32_16x16x4_f32(false, a1, false, b0, (short)0, acc[1][0], false, false);
      acc[1][1] = __builtin_amdgcn_wmma_f32_16x16x4_f32(false, a1, false, b1, (short)0, acc[1][1], false, false);
    }
  }

  const int isDone = *done;
  const float scale = 2.0f / (float)BSZ;
  float lp = 0.f, dd = 0.f;
#pragma unroll
  for (int i = 0; i < 2; ++i)
#pragma unroll
    for (int j = 0; j < 2; ++j)
#pragma unroll
      for (int v = 0; v < 8; ++v) {
        const int gm = m0 + wm * 32 + i * 16 + v + hl * 8;
        const int gn = n0 + wn * 32 + j * 16 + lr;
        const size_t idx = (size_t)gm * HDIM + gn;
        const float rv = r[idx];
        const float pr = rv - bvec[gn];
        lp += pr * pr;                       // loss uses pre-update r
        if (!isDone) {
          const float g  = scale * (pr - acc[i][j][v]);
          const float m1 = 0.9f   * mst[idx] + 0.1f   * g;
          const float v1 = 0.999f * vst[idx] + 0.001f * (g * g);
          mst[idx] = m1;
          vst[idx] = v1;
          const float dr = -0.01f * (m1 * inv1) / (sqrtf(v1 * inv2) + 1e-8f);
          r[idx] = rv + dr;
          dd += dr * dr;
        }
      }

  red[tid] = lp;
  __syncthreads();
  for (int s = 128; s > 0; s >>= 1) {
    if (tid < s) red[tid] += red[tid + s];
    __syncthreads();
  }
  if (tid == 0) P2[blockIdx.y * gridDim.x + blockIdx.x] = red[0];
  __syncthreads();
  red[tid] = dd;
  __syncthreads();
  for (int s = 128; s > 0; s >>= 1) {
    if (tid < s) red[tid] += red[tid + s];
    __syncthreads();
  }
  if (tid == 0) P3[blockIdx.y * gridDim.x + blockIdx.x] = red[0];
}

// ---------------------------------------------------------------------------
// Kernel 3: deterministic final reduction -> losses[t]; convergence flag.
// ---------------------------------------------------------------------------
__global__ __launch_bounds__(256) void loss_reduce_kernel(
    const float* __restrict__ P1, const float* __restrict__ P2,
    const float* __restrict__ P3, float* __restrict__ loss_out,
    int* __restrict__ done, int nb1, int nb2)
{
  __shared__ float red[256];
  const int tid = threadIdx.x;
  float sl = (tid < nb1 ? P1[tid] : 0.f) + (tid < nb2 ? P2[tid] : 0.f);
  float sd = (tid < nb2 ? P3[tid] : 0.f);

  red[tid] = sl;
  __syncthreads();
  for (int s = 128; s > 0; s >>= 1) {
    if (tid < s) red[tid] += red[tid + s];
    __syncthreads();
  }
  if (tid == 0) loss_out[0] = red[0] * (1.0f / (float)BSZ);
  __syncthreads();
  red[tid] = sd;
  __syncthreads();
  for (int s = 128; s > 0; s >>= 1) {
    if (tid < s) red[tid] += red[tid + s];
    __syncthreads();
  }
  if (tid == 0) {
    if (sqrtf(red[0]) < 1e-6f) *done = 1;  // freeze takes effect next iteration
  }
}

// ---------------------------------------------------------------------------
extern "C" void kernel_launch(void* const* d_in, const int* in_sizes, int n_in,
                              void* d_out, int out_size, void* d_ws, size_t ws_size,
                              hipStream_t stream) {
  (void)in_sizes; (void)n_in; (void)out_size; (void)ws_size;
  const float* x  = (const float*)d_in[0];   // (512, 4096)
  const float* W  = (const float*)d_in[1];   // (4096, 2048)
  const float* c  = (const float*)d_in[2];   // (4096,)
  const float* b  = (const float*)d_in[3];   // (2048,)
  const float* r0 = (const float*)d_in[4];   // (512, 2048)
  float* out = (float*)d_out;                // r (512*2048) ++ losses (100)

  const size_t RN = (size_t)BSZ * HDIM;      // 1,048,576
  const size_t DN = (size_t)BSZ * DDIM;      // 2,097,152

  float* ws   = (float*)d_ws;
  float* r    = ws;  ws += RN;
  float* mst  = ws;  ws += RN;
  float* vst  = ws;  ws += RN;
  float* diff = ws;  ws += DN;
  float* P1   = ws;  ws += 256;
  float* P2   = ws;  ws += 128;
  float* P3   = ws;  ws += 128;
  int*   done = (int*)ws;

  hipMemcpyAsync(r, r0, RN * sizeof(float), hipMemcpyDeviceToDevice, stream);
  hipMemsetAsync(mst, 0, 2 * RN * sizeof(float), stream);  // m and v (contiguous)
  hipMemsetAsync(done, 0, sizeof(int), stream);

  float* losses = out + RN;
  const dim3 g1(DDIM / 128, BSZ / 64);   // 32 x 8 = 256 blocks
  const dim3 g2(HDIM / 128, BSZ / 64);   // 16 x 8 = 128 blocks

  for (int t = 0; t < 100; ++t) {
    fwd_kernel<<<g1, 256, 0, stream>>>(x, W, c, r, diff, P1);
    const double t1 = (double)(t + 1);
    const float inv1 = (float)(1.0 / (1.0 - pow(0.9,   t1)));
    const float inv2 = (float)(1.0 / (1.0 - pow(0.999, t1)));
    grad_adam_kernel<<<g2, 256, 0, stream>>>(diff, W, b, r, mst, vst, P2, P3,
                                             done, inv1, inv2);
    loss_reduce_kernel<<<1, 256, 0, stream>>>(P1, P2, P3, losses + t, done, 256, 128);
  }

  hipMemcpyAsync(d_out, r, RN * sizeof(float), hipMemcpyDeviceToDevice, stream);
}